// DelayedSynapticLayer_34737695490052
// MI455X (gfx1250) — compile-verified
//
#include <hip/hip_runtime.h>
#include <hip/hip_bf16.h>
#include <cstdint>

// DelayedSynapticLayer for MI455X (gfx1250).
// out[b,q] = sum_p w[p,q] * lerp(buf[b,tf,p], buf[b,tf+1,p], alpha(p,q))
// Memory-bound (~40MB @ 23.3TB/s -> ~1.7us floor). LDS-staged gather with
// CDNA5 async global->LDS copies; deterministic two-phase reduction via d_ws.
// WMMA deliberately not used: the gather index depends on (p,q), and any
// matrix formulation inflates FLOPs ~25x through the T dimension using the
// slowest WMMA shape (f32 16x16x4) on a problem with ~10 FLOP/byte.

#define B_    16
#define T_    51
#define P_    2048
#define Q_    2048
#define DMAXF 50.0f

#define QT      256                 // q per block == blockDim.x
#define PT      32                  // p per block chunk
#define NCHUNK  (P_ / PT)           // 64 partial chunks
#define TPAD    52                  // padded T stride (dwords)
#define PSTRIDE (B_ * TPAD + 1)     // 833 dwords (odd -> conflict-free staging)
#define SLAB_DW (PT * PSTRIDE)      // 26656 dwords = 106624 B of LDS
#define NPAIR   (B_ * T_)           // 816 (b,t) pairs
#define NSTEP   (NPAIR / 8)         // 102 staging steps per wave (exact)

#if defined(__HIP_DEVICE_COMPILE__) && defined(__gfx1250__) && \
    __has_builtin(__builtin_amdgcn_global_load_async_to_lds_b32)
#define HAVE_ASYNC_LDS 1
#else
#define HAVE_ASYNC_LDS 0
#endif

#if HAVE_ASYNC_LDS
#if __has_builtin(__builtin_amdgcn_s_wait_asynccnt)
#define WAIT_ASYNC0() __builtin_amdgcn_s_wait_asynccnt(0)
#else
#define WAIT_ASYNC0() asm volatile("s_wait_asynccnt 0x0" ::: "memory")
#endif
typedef __attribute__((address_space(1))) int* gas_ptr;   // global int*
typedef __attribute__((address_space(3))) int* las_ptr;   // LDS int*
__device__ __forceinline__ gas_ptr as_global(const void* p) {
  return (gas_ptr)(unsigned long long)(uintptr_t)p;
}
__device__ __forceinline__ las_ptr as_lds(const void* p) {
  // generic LDS address: aperture in high 32 bits, LDS offset in low 32 bits
  return (las_ptr)(unsigned)(uintptr_t)p;
}
#endif

__global__ __launch_bounds__(QT)
void DelayedSynapticLayer_main(const float* __restrict__ buf,
                               const float* __restrict__ weight,
                               const float* __restrict__ draw,
                               float* __restrict__ ws) {
  __shared__ float slab[SLAB_DW];

  const int tid  = threadIdx.x;
  const int lane = tid & 31;          // == p_local during staging
  const int wv   = tid >> 5;          // wave id (8 waves)
  const int q    = blockIdx.x * QT + tid;
  const int p0   = blockIdx.y * PT;

  // ---- Stage buf[:, :, p0 : p0+PT] into LDS, transposed to [p][b][t] ----
  // global element for (pair=(b,t), lane=p_local): buf[pair*P + p0 + lane]
  // lds dword: lane*PSTRIDE + b*TPAD + t   (conflict-free: stride 833 % 64 == 1)
  // Fixed trip count (816/8 == 102) so the compiler needs no divergence
  // bookkeeping around the async-issue loop.
  const float* gsrc = buf + p0 + lane;
#pragma unroll 2
  for (int i = 0; i < NSTEP; ++i) {
    const int pair = wv + (i << 3);
    const int b = pair / T_;
    const int t = pair - b * T_;
    const int ldsdw = lane * PSTRIDE + b * TPAD + t;
#if HAVE_ASYNC_LDS
    __builtin_amdgcn_global_load_async_to_lds_b32(
        as_global(gsrc + (size_t)pair * P_), as_lds(&slab[ldsdw]),
        /*offset=*/0, /*cpol=*/0);
#else
    slab[ldsdw] = gsrc[(size_t)pair * P_];
#endif
  }
#if HAVE_ASYNC_LDS
  WAIT_ASYNC0();
#endif
  __syncthreads();

  // ---- Main loop: stream weight/delay once, gather taps from LDS ----
  float acc[B_];
#pragma unroll
  for (int b = 0; b < B_; ++b) acc[b] = 0.0f;

  const float* wp = weight + (size_t)p0 * Q_ + q;
  const float* dp = draw   + (size_t)p0 * Q_ + q;

  for (int pl = 0; pl < PT; ++pl) {
    // Speculative prefetch (TH=0): silently dropped if past the mapping,
    // so no guard branch needed in the hot loop.
    __builtin_prefetch(wp + (size_t)(pl + 2) * Q_, 0, 0);
    __builtin_prefetch(dp + (size_t)(pl + 2) * Q_, 0, 0);

    const float w = __builtin_nontemporal_load(wp + (size_t)pl * Q_);
    const float x = __builtin_nontemporal_load(dp + (size_t)pl * Q_);

    // d = 50 * sigmoid(x); tf = clip(floor(d), 0, 49); tc = tf+1; alpha = d-tf
    const float d  = DMAXF / (1.0f + __expf(-x));
    float tf = floorf(d);
    tf = fminf(fmaxf(tf, 0.0f), DMAXF - 1.0f);
    const int   ti    = (int)tf;
    const float alpha = d - tf;
    const float wc = w * alpha;       // weight on ceil tap
    const float wf = w - wc;          // weight on floor tap

    // base for (pl, ti); per-b offsets are compile-time DS immediates, and
    // the two taps are adjacent dwords -> pairs into 16 ds ops per iteration
    const float* base = &slab[pl * PSTRIDE + ti];
#pragma unroll
    for (int b = 0; b < B_; ++b) {
      const float vf = base[b * TPAD];       // buf[b, tf,   p]
      const float vc = base[b * TPAD + 1];   // buf[b, tf+1, p]
      acc[b] = fmaf(wf, vf, acc[b]);
      acc[b] = fmaf(wc, vc, acc[b]);
    }
  }

  // ---- Write partials: ws[chunk][b][q], coalesced ----
  float* wsout = ws + ((size_t)blockIdx.y * B_) * Q_ + q;
#pragma unroll
  for (int b = 0; b < B_; ++b) wsout[(size_t)b * Q_] = acc[b];
}

// Deterministic reduction of NCHUNK partials -> out[b,q]
__global__ __launch_bounds__(256)
void DelayedSynapticLayer_reduce(const float* __restrict__ ws,
                                 float* __restrict__ out) {
  const int idx = blockIdx.x * 256 + threadIdx.x;   // 0 .. B_*Q_-1
  float s = 0.0f;
  for (int c = 0; c < NCHUNK; ++c) {
    s += ws[(size_t)c * (B_ * Q_) + idx];
  }
  out[idx] = s;
}

extern "C" void kernel_launch(void* const* d_in, const int* in_sizes, int n_in,
                              void* d_out, int out_size, void* d_ws, size_t ws_size,
                              hipStream_t stream) {
  const float* buf    = (const float*)d_in[0];   // [16, 51, 2048]
  const float* weight = (const float*)d_in[1];   // [2048, 2048]
  const float* draw   = (const float*)d_in[2];   // [2048, 2048]
  float*       out    = (float*)d_out;           // [16, 2048]
  float*       ws     = (float*)d_ws;            // NCHUNK*16*2048 f32 = 8 MB

  dim3 gridA(Q_ / QT, NCHUNK);                   // 8 x 64 = 512 blocks
  DelayedSynapticLayer_main<<<gridA, QT, 0, stream>>>(buf, weight, draw, ws);

  dim3 gridB((B_ * Q_) / 256);                   // 128 blocks
  DelayedSynapticLayer_reduce<<<gridB, 256, 0, stream>>>(ws, out);
}